// LargeLBCS_46823733461848
// MI455X (gfx1250) — compile-verified
//
#include <hip/hip_runtime.h>
#include <cstddef>

// ---------------------------------------------------------------------------
// LargeLBCS loss on MI455X (gfx1250, wave32).
//
// sc[n,s,q] = sum_p pauli[n,q,p] * h[s,q,p]   (K=3 GEMM -> WMMA 16x16x4 f32)
// cov[n,s]  = prod_q sc[n,s,q]                (folded via v_pk_mul_f32)
// loss      = sum_n coeff[n]^2 / (sum_s r[s]*cov[n,s])
//
// K=4 zero-padded fragment layouts: all 32 lanes load unconditionally
// (hi half reads the zeroed K=3 slot) -> no exec-masked divergent loads.
// A from LDS via ds_load_b64 (stride 204: aligned + bank-conflict-free),
// B from workspace as coalesced global_load_b64, double-buffered across q.
// ---------------------------------------------------------------------------

typedef float v2f __attribute__((ext_vector_type(2)));
typedef float v8f __attribute__((ext_vector_type(8)));

static constexpr int S_DIM  = 128;         // s (heads)
static constexpr int Q_DIM  = 50;          // q
static constexpr int QP     = Q_DIM * 3;   // 150 floats per row
static constexpr int LROW   = 204;         // LDS row stride (dwords): Q*4 padded to mult-of-4
static constexpr int NTILES = S_DIM / 16;  // 8 s-tiles

__device__ __forceinline__ float softplus20(float x) {
    float y = 20.0f * x;
    return (y > 30.0f) ? y : log1pf(expf(y));
}

// --- Kernel 1: r activation + zero loss slot --------------------------------
__global__ void lbcs_act_r(const float* __restrict__ head_ratios,
                           float* __restrict__ out, int S) {
    __shared__ float sbuf[256];
    __shared__ float ssum;
    int i = threadIdx.x;
    float rv = 0.0f;
    if (i < S) rv = softplus20(head_ratios[i]) + 0.001f / (float)S / 1.001f;
    sbuf[i] = rv;
    __syncthreads();
    if (i == 0) {
        float s = 0.0f;
        for (int j = 0; j < S; ++j) s += fabsf(sbuf[j]);
        ssum = fmaxf(s, 1e-12f);
        out[0] = 0.0f;                     // loss accumulator
    }
    __syncthreads();
    if (i < S) out[1 + i] = rv / ssum;
}

// --- Kernel 2: h activation -> d_out, plus paired/padded copy in ws ---------
// ws layout: v2f hT2[(q*2 + half)][s];  half0 = {k0,k1}, half1 = {k2, 0}.
__global__ void lbcs_act_h(const float* __restrict__ heads,
                           float* __restrict__ h_out,      // [s][q][p]
                           v2f* __restrict__ hT2,          // ws (may be null)
                           int use_ws, int S, int Q) {
    int idx = blockIdx.x * blockDim.x + threadIdx.x;        // (s,q) pair
    if (idx >= S * Q) return;
    int s = idx / Q, q = idx % Q;
    const float* src = heads + (size_t)idx * 3;
    float a = softplus20(src[0]);
    float b = softplus20(src[1]);
    float c = softplus20(src[2]);
    float inv = 1.0f / fmaxf(a + b + c, 1e-12f);            // values >= 0
    a *= inv; b *= inv; c *= inv;
    float* dst = h_out + (size_t)idx * 3;
    dst[0] = a; dst[1] = b; dst[2] = c;
    if (use_ws) {
        v2f lo; lo.x = a; lo.y = b;
        v2f hi; hi.x = c; hi.y = 0.0f;
        hT2[(size_t)(q * 2 + 0) * S + s] = lo;
        hT2[(size_t)(q * 2 + 1) * S + s] = hi;
    }
}

// B-fragment load: one aligned 64-bit load (ws path) or guarded fallback.
__device__ __forceinline__ v2f load_bfrag(const v2f* __restrict__ bbase,
                                          const float* __restrict__ h_flat,
                                          int use_ws, int q, int scol,
                                          int k0, bool hi) {
    if (use_ws) {
        return bbase[(size_t)(q * 2) * S_DIM + scol];       // bbase includes half+col
    } else {
        const float* hp = h_flat + (size_t)scol * QP + q * 3;
        v2f b;
        b.x = hp[k0];
        b.y = hi ? 0.0f : hp[1];
        return b;
    }
}

// --- Kernel 3: WMMA main loop -----------------------------------------------
// Block = 128 threads = 4 waves; each wave owns 16 consecutive n-rows x S=128.
__global__ void __launch_bounds__(128)
lbcs_main(const float* __restrict__ pauli,     // [n][q][p]
          const float* __restrict__ coeff,     // [n]
          const float* __restrict__ r_norm,    // d_out + 1
          const float* __restrict__ h_flat,    // d_out + 1 + S (fallback B src)
          const v2f*   __restrict__ hT2,       // ws paired h (preferred)
          int use_ws,
          float* __restrict__ loss_out,        // d_out + 0
          int N) {
    __shared__ float lA[4][16 * LROW];

    const int lane = threadIdx.x & 31;
    const int wave = threadIdx.x >> 5;
    const int n0   = (blockIdx.x * 4 + wave) * 16;
    const bool active = (n0 < N);              // wave-uniform

    if (active) {
        float* la = lA[wave];
        // Zero the K=3 pad slots so hi-half fragment loads read 0.
        for (int i = lane; i < 16 * Q_DIM; i += 32) {
            int row = i / Q_DIM, q = i - row * Q_DIM;
            la[row * LROW + q * 4 + 3] = 0.0f;
        }
        // Coalesced read of 16 rows x 150 floats, scattered into K=4 layout.
        const float* src = pauli + (size_t)n0 * QP;
        const int rows  = (N - n0 < 16) ? (N - n0) : 16;
        const int limit = rows * QP;
        for (int i = lane; i < 16 * QP; i += 32) {
            int row = i / QP, c = i - row * QP;
            int q = c / 3, k = c - q * 3;
            float v = (i < limit) ? src[i] : 0.0f;
            la[row * LROW + q * 4 + k] = v;
        }
    }
    __syncthreads();

    if (active) {
        const int  col  = lane & 15;           // M (A) / N (B,D) within tile
        const bool hi   = (lane >= 16);        // upper half: K = {2,3}
        const int  k0   = hi ? 2 : 0;
        const float* lrow = &lA[wave][col * LROW];
        // ws B base folds in the K-half row and the lane column.
        const v2f* bbase = hT2 + (size_t)(hi ? 1 : 0) * S_DIM + col;

        const v8f one  = {1.f, 1.f, 1.f, 1.f, 1.f, 1.f, 1.f, 1.f};
        const v8f zero = {0.f, 0.f, 0.f, 0.f, 0.f, 0.f, 0.f, 0.f};
        v8f cov[NTILES];
        #pragma unroll
        for (int t = 0; t < NTILES; ++t) cov[t] = one;

        v2f bufA[NTILES], bufB[NTILES];
        #pragma unroll
        for (int t = 0; t < NTILES; ++t)
            bufA[t] = load_bfrag(bbase, h_flat, use_ws, 0, t * 16 + col, k0, hi);

        // q loop, 2x unrolled (Q even) with double-buffered B fragments.
        for (int q = 0; q < Q_DIM; q += 2) {
            #pragma unroll
            for (int t = 0; t < NTILES; ++t)
                bufB[t] = load_bfrag(bbase, h_flat, use_ws, q + 1, t * 16 + col, k0, hi);

            v2f a0 = *(const v2f*)(lrow + q * 4 + k0);       // ds_load_b64
            #pragma unroll
            for (int t = 0; t < NTILES; ++t) {
                v8f d = __builtin_amdgcn_wmma_f32_16x16x4_f32(
                    false, a0, false, bufA[t], (short)0, zero, false, false);
                cov[t] *= d;
            }

            if (q + 2 < Q_DIM) {
                #pragma unroll
                for (int t = 0; t < NTILES; ++t)
                    bufA[t] = load_bfrag(bbase, h_flat, use_ws, q + 2, t * 16 + col, k0, hi);
            }

            v2f a1 = *(const v2f*)(lrow + (q + 1) * 4 + k0);
            #pragma unroll
            for (int t = 0; t < NTILES; ++t) {
                v8f d = __builtin_amdgcn_wmma_f32_16x16x4_f32(
                    false, a1, false, bufB[t], (short)0, zero, false, false);
                cov[t] *= d;
            }
        }

        // r-weighted sum over s, then per-row lane reduction.
        float rv[NTILES];
        #pragma unroll
        for (int t = 0; t < NTILES; ++t) rv[t] = r_norm[t * 16 + col];

        float acc[8];
        #pragma unroll
        for (int v = 0; v < 8; ++v) {
            float s = 0.0f;
            #pragma unroll
            for (int t = 0; t < NTILES; ++t) s += cov[t][v] * rv[t];
            #pragma unroll
            for (int off = 1; off < 16; off <<= 1)   // reduce each 16-lane half
                s += __shfl_xor(s, off, 32);
            acc[v] = s;
        }

        if (lane == 0 || lane == 16) {     // lane0: rows v, lane16: rows v+8
            const int base = n0 + (hi ? 8 : 0);
            float contrib = 0.0f;
            #pragma unroll
            for (int v = 0; v < 8; ++v) {
                int n = base + v;
                if (n < N) {
                    float c = coeff[n];
                    contrib += c * c / acc[v];
                }
            }
            atomicAdd(loss_out, contrib);
        }
    }
}

// ---------------------------------------------------------------------------
extern "C" void kernel_launch(void* const* d_in, const int* in_sizes, int n_in,
                              void* d_out, int out_size, void* d_ws, size_t ws_size,
                              hipStream_t stream) {
    const float* heads  = (const float*)d_in[0];   // (s,q,p)
    const float* ratios = (const float*)d_in[1];   // (s,)
    const float* pauli  = (const float*)d_in[2];   // (n,q,p)
    const float* coeff  = (const float*)d_in[3];   // (n,)
    float* out = (float*)d_out;                    // [loss | r(S) | h(S*Q*3)]

    const int S = in_sizes[1];                     // 128
    const int Q = in_sizes[0] / (S * 3);           // 50
    const int N = in_sizes[3];                     // 20000

    // Paired/padded h in workspace if it fits (one b64 load per B fragment).
    const size_t need = (size_t)Q * 2 * S * sizeof(v2f);   // 102,400 B
    v2f* hT2 = nullptr;
    int use_ws = 0;
    if (ws_size >= need) { hT2 = (v2f*)d_ws; use_ws = 1; }

    lbcs_act_r<<<1, S, 0, stream>>>(ratios, out, S);

    const int totalSQ = S * Q;
    lbcs_act_h<<<(totalSQ + 255) / 256, 256, 0, stream>>>(
        heads, out + 1 + S, hT2, use_ws, S, Q);

    const int waves  = (N + 15) / 16;
    const int blocks = (waves + 3) / 4;
    lbcs_main<<<blocks, 128, 0, stream>>>(
        pauli, coeff, out + 1, out + 1 + S, hT2, use_ws, out, N);
}